// GNNModelWithStylesAndEdges_85753317032235
// MI455X (gfx1250) — compile-verified
//
#include <hip/hip_runtime.h>
#include <cmath>

#define NA 50000
#define NS 5000
#define E_AA 500000
#define E_AS 250000
#define C 128
#define ED 8

typedef __attribute__((ext_vector_type(2))) float v2f;
typedef __attribute__((ext_vector_type(8))) float v8f;

typedef __attribute__((address_space(3))) float lds_float;

__global__ void fill_kernel(float* __restrict__ p, float v, int n) {
    int i = blockIdx.x * blockDim.x + threadIdx.x;
    if (i < n) p[i] = v;
}

// Y[N,128] = X[N,K] @ W[K,128], row-major, fp32 WMMA 16x16x4.
// Block = 256 threads (8 waves). Block owns 32 rows (two 16-row tiles per wave);
// wave w owns column tile n0=16*w. A tile staged to LDS via async global->LDS
// (row stride K+4 dwords -> conflict-free ds_load_b64). K must be 64 or 128.
__global__ void gemm_wmma_kernel(const float* __restrict__ X,
                                 const float* __restrict__ W,
                                 float* __restrict__ Y, int N, int K) {
    __shared__ float As[32 * 132];                 // 32 rows x (128+4) dwords max
    const int lane = threadIdx.x & 31;
    const int wave = threadIdx.x >> 5;
    const int m0 = blockIdx.x * 32;
    const int n0 = wave * 16;
    const int lo = lane & 15;                      // M (A/D) or N (B/D)
    const int hi = lane >> 4;                      // K-pair select / M half
    const int stride = K + 4;
    const int kshift = (K == 128) ? 7 : 6;

    // ---- async stage A tile (32 x K), rows clamped to N-1 ----
    {
        const unsigned lds_base = (unsigned)(uintptr_t)(lds_float*)As;
        const int elems = 32 * K;
        for (int e = threadIdx.x; e < elems; e += blockDim.x) {
            int r = e >> kshift;
            int k = e & (K - 1);
            int row = m0 + r;
            if (row >= N) row = N - 1;
            unsigned lds_addr = lds_base + (unsigned)((r * stride + k) * 4);
            unsigned long long gaddr =
                (unsigned long long)(const void*)(X + (size_t)row * K + k);
            asm volatile("global_load_async_to_lds_b32 %0, %1, off"
                         :: "v"(lds_addr), "v"(gaddr) : "memory");
        }
        asm volatile("s_wait_asynccnt 0" ::: "memory");
        __syncthreads();
    }

    v8f acc0 = {0.f, 0.f, 0.f, 0.f, 0.f, 0.f, 0.f, 0.f};
    v8f acc1 = {0.f, 0.f, 0.f, 0.f, 0.f, 0.f, 0.f, 0.f};
    for (int k0 = 0; k0 < K; k0 += 4) {
        const float* a0p = &As[lo * stride + k0 + 2 * hi];
        v2f a0; a0.x = a0p[0]; a0.y = a0p[1];
        const float* a1p = &As[(lo + 16) * stride + k0 + 2 * hi];
        v2f a1; a1.x = a1p[0]; a1.y = a1p[1];
        const float* bp = W + (size_t)(k0 + 2 * hi) * C + n0 + lo;
        v2f b; b.x = bp[0]; b.y = bp[C];
#if __has_builtin(__builtin_amdgcn_wmma_f32_16x16x4_f32)
        acc0 = __builtin_amdgcn_wmma_f32_16x16x4_f32(false, a0, false, b, (short)0, acc0, false, false);
        acc1 = __builtin_amdgcn_wmma_f32_16x16x4_f32(false, a1, false, b, (short)0, acc1, false, false);
#else
        acc0[0] += a0.x * b.x; acc1[0] += a1.x * b.x;
#endif
    }
#pragma unroll
    for (int v = 0; v < 8; ++v) {
        int row0 = m0 + v + 8 * hi;
        if (row0 < N) Y[(size_t)row0 * C + n0 + lo] = acc0[v];
        int row1 = m0 + 16 + v + 8 * hi;
        if (row1 < N) Y[(size_t)row1 * C + n0 + lo] = acc1[v];
    }
}

__device__ __forceinline__ void atomic_max_f32(float* addr, float val) {
    // signed-max / unsigned-min ordering trick, valid with -inf init
    if (val >= 0.f) atomicMax((int*)addr, __float_as_int(val));
    else            atomicMin((unsigned int*)addr, __float_as_uint(val));
}

// Grid-stride, one wave per edge: logit = leaky_relu(xl[src]+xr[dst]+eattr@We, 0.2) . att
// We (8x128) and att hoisted to registers once per wave.
__global__ void edge_logit_kernel(const float* __restrict__ xl, const float* __restrict__ xr,
                                  const float* __restrict__ eattr, const float* __restrict__ We,
                                  const float* __restrict__ att,
                                  const int* __restrict__ src, const int* __restrict__ dst,
                                  float* __restrict__ elog, float* __restrict__ dmax, int E) {
    const int lane = threadIdx.x & 31;
    const int wid = blockIdx.x * (blockDim.x >> 5) + (threadIdx.x >> 5);
    const int nw = gridDim.x * (blockDim.x >> 5);
    float wreg[4 * ED];
    float attv[4];
#pragma unroll
    for (int j = 0; j < 4; ++j) {
        int c = lane + 32 * j;
        attv[j] = att[c];
#pragma unroll
        for (int k = 0; k < ED; ++k) wreg[j * ED + k] = We[k * C + c];
    }
    for (int e = wid; e < E; e += nw) {
        int s = src[e], d = dst[e];
        float ea[ED];
#pragma unroll
        for (int k = 0; k < ED; ++k) ea[k] = eattr[(size_t)e * ED + k];
        float acc = 0.f;
#pragma unroll
        for (int j = 0; j < 4; ++j) {
            int c = lane + 32 * j;
            float m = xl[(size_t)s * C + c] + xr[(size_t)d * C + c];
#pragma unroll
            for (int k = 0; k < ED; ++k) m = fmaf(ea[k], wreg[j * ED + k], m);
            float lr = (m > 0.f) ? m : 0.2f * m;
            acc = fmaf(lr, attv[j], acc);
        }
#pragma unroll
        for (int off = 16; off > 0; off >>= 1) acc += __shfl_xor(acc, off, 32);
        if (lane == 0) {
            elog[e] = acc;
            atomic_max_f32(&dmax[d], acc);
        }
    }
}

// One thread per edge: ee = exp(e - max[dst]); den[dst] += ee
__global__ void edge_exp_kernel(float* __restrict__ elog, const float* __restrict__ dmax,
                                float* __restrict__ den, const int* __restrict__ dst, int E) {
    int e = blockIdx.x * blockDim.x + threadIdx.x;
    if (e >= E) return;
    int d = dst[e];
    float ee = __expf(elog[e] - dmax[d]);
    elog[e] = ee;
    atomicAdd(&den[d], ee);
}

// Grid-stride, one wave per edge: out[dst] += alpha * xl[src]
__global__ void edge_scatter_kernel(const float* __restrict__ elog, const float* __restrict__ den,
                                    const float* __restrict__ xl,
                                    const int* __restrict__ src, const int* __restrict__ dst,
                                    float* __restrict__ out, int E) {
    const int lane = threadIdx.x & 31;
    const int wid = blockIdx.x * (blockDim.x >> 5) + (threadIdx.x >> 5);
    const int nw = gridDim.x * (blockDim.x >> 5);
    for (int e = wid; e < E; e += nw) {
        int s = src[e], d = dst[e];
        float alpha = elog[e] / fmaxf(den[d], 1e-16f);
#pragma unroll
        for (int j = 0; j < 4; ++j) {
            int c = lane + 32 * j;
            atomicAdd(&out[(size_t)d * C + c], alpha * xl[(size_t)s * C + c]);
        }
    }
}

// One wave per row: t = in + b; optional relu; L2 normalize; write out.
__global__ void finalize_kernel(const float* __restrict__ in, const float* __restrict__ bias,
                                float* __restrict__ out, int N, int do_relu) {
    int row = blockIdx.x * (blockDim.x >> 5) + (threadIdx.x >> 5);
    int lane = threadIdx.x & 31;
    if (row >= N) return;
    float t[4];
    float ss = 0.f;
#pragma unroll
    for (int j = 0; j < 4; ++j) {
        int c = lane + 32 * j;
        float v = in[(size_t)row * C + c] + bias[c];
        if (do_relu) v = fmaxf(v, 0.f);
        t[j] = v;
        ss += v * v;
    }
#pragma unroll
    for (int off = 16; off > 0; off >>= 1) ss += __shfl_xor(ss, off, 32);
    float inv = 1.0f / fmaxf(sqrtf(ss), 1e-12f);
#pragma unroll
    for (int j = 0; j < 4; ++j) out[(size_t)row * C + lane + 32 * j] = t[j] * inv;
}

static inline int cdiv(int a, int b) { return (a + b - 1) / b; }

extern "C" void kernel_launch(void* const* d_in, const int* in_sizes, int n_in,
                              void* d_out, int out_size, void* d_ws, size_t ws_size,
                              hipStream_t stream) {
    (void)in_sizes; (void)n_in; (void)out_size; (void)ws_size;

    const float* x_artist = (const float*)d_in[0];
    const float* x_style  = (const float*)d_in[1];
    const int*   src_aa   = (const int*)d_in[2];
    const int*   dst_aa   = (const int*)d_in[3];
    const float* eattr_aa = (const float*)d_in[4];
    const int*   src_as   = (const int*)d_in[5];
    const int*   dst_as   = (const int*)d_in[6];
    const float* eattr_as = (const float*)d_in[7];

    // workspace layout (floats)
    float* ws   = (float*)d_ws;
    float* ha   = ws;                       // NA*C
    float* hs   = ha + (size_t)NA * C;      // NS*C
    float* na   = hs + (size_t)NS * C;      // NA*C
    float* ns   = na + (size_t)NA * C;      // NS*C
    float* xl   = ns + (size_t)NS * C;      // NA*C
    float* xr   = xl + (size_t)NA * C;      // NA*C (max of NA,NS)
    float* ebuf = xr + (size_t)NA * C;      // E_AA (max of E_AA,E_AS)
    float* dmax = ebuf + E_AA;              // NA (max of NA,NS)
    float* den  = dmax + NA;                // NA

    float* out_final = (float*)d_out;

    for (int i = 0; i < 3; ++i) {
        const float *WlA, *WrA, *attA, *WeA, *bA;
        const float *WlS, *WrS, *attS, *WeS, *bS;
        const float *inA, *inS;
        int Ka, Ks;
        if (i == 0) {
            inA = x_artist; inS = x_style; Ka = 128; Ks = 64;
            WlA = (const float*)d_in[8];  WrA = (const float*)d_in[9];
            attA = (const float*)d_in[10]; WeA = (const float*)d_in[11]; bA = (const float*)d_in[12];
            WlS = (const float*)d_in[13]; WrS = (const float*)d_in[14];
            attS = (const float*)d_in[15]; WeS = (const float*)d_in[16]; bS = (const float*)d_in[17];
        } else {
            int j = i - 1;
            inA = ha; inS = hs; Ka = C; Ks = C;
            WlA  = (const float*)d_in[18] + (size_t)j * C * C;
            WrA  = (const float*)d_in[19] + (size_t)j * C * C;
            attA = (const float*)d_in[20] + (size_t)j * C;
            WeA  = (const float*)d_in[21] + (size_t)j * ED * C;
            bA   = (const float*)d_in[22] + (size_t)j * C;
            WlS  = (const float*)d_in[23] + (size_t)j * C * C;
            WrS  = (const float*)d_in[24] + (size_t)j * C * C;
            attS = (const float*)d_in[25] + (size_t)j * C;
            WeS  = (const float*)d_in[26] + (size_t)j * ED * C;
            bS   = (const float*)d_in[27] + (size_t)j * C;
        }

        // ---- edge type aa (artist -> artist, NA destinations) ----
        gemm_wmma_kernel<<<cdiv(NA, 32), 256, 0, stream>>>(inA, WlA, xl, NA, Ka);
        gemm_wmma_kernel<<<cdiv(NA, 32), 256, 0, stream>>>(inA, WrA, xr, NA, Ka);
        fill_kernel<<<cdiv(NA, 256), 256, 0, stream>>>(dmax, -INFINITY, NA);
        fill_kernel<<<cdiv(NA, 256), 256, 0, stream>>>(den, 0.f, NA);
        fill_kernel<<<cdiv(NA * C, 256), 256, 0, stream>>>(na, 0.f, NA * C);
        edge_logit_kernel<<<cdiv(E_AA, 8 * 16), 256, 0, stream>>>(xl, xr, eattr_aa, WeA, attA,
                                                                  src_aa, dst_aa, ebuf, dmax, E_AA);
        edge_exp_kernel<<<cdiv(E_AA, 256), 256, 0, stream>>>(ebuf, dmax, den, dst_aa, E_AA);
        edge_scatter_kernel<<<cdiv(E_AA, 8 * 16), 256, 0, stream>>>(ebuf, den, xl, src_aa, dst_aa, na, E_AA);

        // ---- edge type as (artist -> style, NS destinations) ----
        gemm_wmma_kernel<<<cdiv(NA, 32), 256, 0, stream>>>(inA, WlS, xl, NA, Ka);
        gemm_wmma_kernel<<<cdiv(NS, 32), 256, 0, stream>>>(inS, WrS, xr, NS, Ks);
        fill_kernel<<<cdiv(NS, 256), 256, 0, stream>>>(dmax, -INFINITY, NS);
        fill_kernel<<<cdiv(NS, 256), 256, 0, stream>>>(den, 0.f, NS);
        fill_kernel<<<cdiv(NS * C, 256), 256, 0, stream>>>(ns, 0.f, NS * C);
        edge_logit_kernel<<<cdiv(E_AS, 8 * 16), 256, 0, stream>>>(xl, xr, eattr_as, WeS, attS,
                                                                  src_as, dst_as, ebuf, dmax, E_AS);
        edge_exp_kernel<<<cdiv(E_AS, 256), 256, 0, stream>>>(ebuf, dmax, den, dst_as, E_AS);
        edge_scatter_kernel<<<cdiv(E_AS, 8 * 16), 256, 0, stream>>>(ebuf, den, xl, src_as, dst_as, ns, E_AS);

        // ---- bias + relu + l2norm ----
        int do_relu = (i < 2) ? 1 : 0;
        float* outA = (i == 2) ? out_final : ha;
        float* outS = (i == 2) ? (out_final + (size_t)NA * C) : hs;
        finalize_kernel<<<cdiv(NA, 8), 256, 0, stream>>>(na, bA, outA, NA, do_relu);
        finalize_kernel<<<cdiv(NS, 8), 256, 0, stream>>>(ns, bS, outS, NS, do_relu);
    }
}